// CTCBeamSearchDecoder_19361712570816
// MI455X (gfx1250) — compile-verified
//
#include <hip/hip_runtime.h>
#include <hip/hip_bf16.h>

#define TT 512
#define VV 96
#define CC 95
#define WW 16
#define NEGF (-1e30f)

typedef float v2f __attribute__((ext_vector_type(2)));
typedef float v8f __attribute__((ext_vector_type(8)));

__device__ __forceinline__ float laddexp(float a, float b) {
    float m = fmaxf(a, b), n = fminf(a, b);
    return m + log1pf(__expf(n - m));
}

__global__ __launch_bounds__(32)
void ctc_beam_kernel(const float* __restrict__ in, int* __restrict__ dec,
                     float* __restrict__ prob) {
    __shared__ float lp[VV];                 // log-softmax row for this timestep
    __shared__ unsigned short bt[TT * WW];   // backtrace: stay<<15 | parent<<8 | char

    const int b    = blockIdx.x;
    const int lane = threadIdx.x;
    const float* base = in + (size_t)b * TT * VV;

    // Beam state lives in lanes 0..15 (beam == lane).
    float p_b  = (lane == 0) ? 0.f : NEGF;
    float p_nb = NEGF;
    int   len  = 0, last = -1;
    const int hi  = (lane >= 16) ? 8 : 0;
    const int col = lane & 15;

    for (int t = 0; t < TT; t++) {
        // ---- log-softmax of row (b, t): 96 values, 3 per lane ----
        const float* row = base + t * VV;
        float x0 = row[lane], x1 = row[lane + 32], x2 = row[lane + 64];
        if (t + 1 < TT) __builtin_prefetch(row + VV, 0, 1);  // global_prefetch_b8
        float m = fmaxf(x0, fmaxf(x1, x2));
        #pragma unroll
        for (int o = 16; o; o >>= 1) m = fmaxf(m, __shfl_xor(m, o, 32));
        float s = __expf(x0 - m) + __expf(x1 - m) + __expf(x2 - m);
        #pragma unroll
        for (int o = 16; o; o >>= 1) s += __shfl_xor(s, o, 32);
        float lse = m + __logf(s);
        lp[lane] = x0 - lse; lp[lane + 32] = x1 - lse; lp[lane + 64] = x2 - lse;
        __syncthreads();

        // ---- per-old-beam quantities (meaningful in lanes 0..15) ----
        float lp_blank = lp[CC];
        float ptot = laddexp(p_b, p_nb);
        float npb  = ptot + lp_blank;                        // new_p_b[w]
        float lp_last = lp[last < 0 ? 0 : last];
        float nnbs = (last >= 0) ? (p_nb + lp_last) : NEGF;  // new_p_nb_stay[w]
        float stay_c = (lane < 16) ? laddexp(npb, nnbs) : NEGF;

        // broadcast per-row beam info used by the WMMA fixup
        float pb_r[8]; int last_r[8];
        #pragma unroll
        for (int r = 0; r < 8; r++) {
            pb_r[r]   = __shfl(p_b,  r + hi, 32);
            last_r[r] = __shfl(last, r + hi, 32);
        }

        // ---- ext scores via V_WMMA_F32_16X16X4_F32: D = [ptot|1] x [1;lp] ----
        // A 16x4 f32: lanes 0-15 hold K=0 (VGPR0) and K=1 (VGPR1); lanes 16-31 (K=2,3) = 0
        v2f A; A.x = (lane < 16) ? ptot : 0.f; A.y = (lane < 16) ? 1.f : 0.f;
        float cand[48];
        #pragma unroll
        for (int tile = 0; tile < 6; tile++) {
            int c = tile * 16 + col;
            float lpc = (c == CC) ? NEGF : lp[c];   // kill the blank column
            // B 4x16 f32: row K=0 = ones, row K=1 = lp[c]; rows 2,3 (lanes 16-31) = 0
            v2f Bm; Bm.x = (lane < 16) ? 1.f : 0.f; Bm.y = (lane < 16) ? lpc : 0.f;
            v8f cz = {0.f, 0.f, 0.f, 0.f, 0.f, 0.f, 0.f, 0.f};
            v8f D = __builtin_amdgcn_wmma_f32_16x16x4_f32(
                false, A, false, Bm, (short)0, cz, false, false);
            #pragma unroll
            for (int r = 0; r < 8; r++) {           // VGPR r: beam r (lanes 0-15) / r+8 (16-31)
                float v = D[r];
                if (c == last_r[r]) v = pb_r[r] + lpc;   // "same" -> base = p_b
                cand[tile * 8 + r] = v;
            }
        }

        // ---- top-16 of 16 stay + 16x95 ext candidates: iterative wave argmax ----
        float selS = NEGF; int selCode = 0;
        #pragma unroll 1
        for (int k = 0; k < WW; k++) {
            float bv = stay_c; int bs = 48;
            #pragma unroll
            for (int s2 = 0; s2 < 48; s2++) if (cand[s2] > bv) { bv = cand[s2]; bs = s2; }
            int code = (lane << 6) | bs;
            #pragma unroll
            for (int o = 16; o; o >>= 1) {
                float ov = __shfl_xor(bv, o, 32);
                int   oc = __shfl_xor(code, o, 32);
                if (ov > bv || (ov == bv && oc < code)) { bv = ov; code = oc; }
            }
            if (lane == k) { selS = bv; selCode = code; }
            int wl = code >> 6, ws = code & 63;
            if (lane == wl) {
                if (ws == 48) stay_c = NEGF;
                else {
                    #pragma unroll
                    for (int s2 = 0; s2 < 48; s2++) if (s2 == ws) cand[s2] = NEGF;
                }
            }
        }

        // ---- new beam j = lane (lanes 0..15): decode winner, gather parent state ----
        int  wl   = selCode >> 6, ws = selCode & 63;
        bool stay = (ws == 48);
        int  w    = stay ? wl : ((ws & 7) + ((wl >= 16) ? 8 : 0));
        int  ch   = stay ? -1 : ((ws >> 3) * 16 + (wl & 15));
        float o_npb  = __shfl(npb,  w, 32);
        float o_nnbs = __shfl(nnbs, w, 32);
        int   o_len  = __shfl(len,  w, 32);
        int   o_last = __shfl(last, w, 32);
        float np_b  = stay ? o_npb  : NEGF;
        float np_nb = stay ? o_nnbs : selS;
        int   n_len = stay ? o_len  : (o_len + 1);
        int   n_last= stay ? o_last : ch;
        if (lane < 16)
            bt[t * WW + lane] = (unsigned short)((stay ? 0x8000 : 0) | (w << 8) | (ch & 0xFF));
        __syncthreads();
        if (lane < 16) { p_b = np_b; p_nb = np_nb; len = n_len; last = n_last; }
    }

    // ---- final top-1 + backtrace reconstruction ----
    float fs = (lane < 16) ? laddexp(p_b, p_nb) : NEGF;
    int bi = lane;
    #pragma unroll
    for (int o = 16; o; o >>= 1) {
        float ov = __shfl_xor(fs, o, 32); int oi = __shfl_xor(bi, o, 32);
        if (ov > fs || (ov == fs && oi < bi)) { fs = ov; bi = oi; }
    }
    int blen = __shfl(len, bi, 32);
    __syncthreads();
    if (lane == 0) {
        prob[b] = __expf(fs);
        int cur = bi, pos = blen - 1;
        for (int t = TT - 1; t >= 0; t--) {
            unsigned short e = bt[t * WW + cur];
            if (!(e & 0x8000)) {
                int c2 = e & 0xFF;
                if (pos >= 0) dec[(size_t)b * TT + pos] = c2;
                pos--;
            }
            cur = (e >> 8) & 0x7F;
        }
    }
    for (int i = blen + lane; i < TT; i += 32) dec[(size_t)b * TT + i] = -1;
}

extern "C" void kernel_launch(void* const* d_in, const int* in_sizes, int n_in,
                              void* d_out, int out_size, void* d_ws, size_t ws_size,
                              hipStream_t stream) {
    (void)n_in; (void)d_ws; (void)ws_size; (void)out_size;
    const float* in = (const float*)d_in[0];
    int B = in_sizes[0] / (TT * VV);           // 512
    int*   dec  = (int*)d_out;                 // decoded (B,1,T) int32
    float* prob = (float*)d_out + (size_t)B * TT;  // probability (B,1) float32
    ctc_beam_kernel<<<B, 32, 0, stream>>>(in, dec, prob);
}